// GaussRenderer_87101936762902
// MI455X (gfx1250) — compile-verified
//
#include <hip/hip_runtime.h>
#include <math.h>

#define N_G    1024
#define W_IMG  200
#define H_IMG  200
#define TILE_SZ 25
#define NPIX   625      // TILE_SZ*TILE_SZ
#define MPAD   640      // 40 M-tiles of 16 rows
#define NCHUNK (N_G/16)

#define TANF_C  0.5773502691896258f
#define FOCAL_C 173.20508075688772f

typedef __attribute__((ext_vector_type(2))) float v2f;
typedef __attribute__((ext_vector_type(8))) float v8f;

__device__ __forceinline__ float clampf(float v, float lo, float hi) {
  return fminf(fmaxf(v, lo), hi);
}

// ---------------------------------------------------------------------------
// Kernel 1: per-Gaussian preprocessing.
// Produces: depth, 8 quadratic-form coefficients (K-padded), opacity, color,
// clipped bbox, and writes radii straight to the output buffer.
// ---------------------------------------------------------------------------
__global__ void gs_preprocess(const float* __restrict__ means,
                              const float* __restrict__ scales,
                              const float* __restrict__ rots,
                              const float* __restrict__ opac,
                              const float* __restrict__ cols,
                              const float* __restrict__ vm,
                              const float* __restrict__ pm,
                              float* __restrict__ depth,
                              float* __restrict__ coefU,
                              float* __restrict__ opU,
                              float* __restrict__ colU,
                              float* __restrict__ rminU,
                              float* __restrict__ rmaxU,
                              float* __restrict__ radiiOut)
{
  int i = blockIdx.x * blockDim.x + threadIdx.x;
  if (i >= N_G) return;

  // --- quaternion -> rotation ---
  float qr = rots[i*4+0], qx = rots[i*4+1], qy = rots[i*4+2], qz = rots[i*4+3];
  float qinv = 1.0f / sqrtf(qr*qr + qx*qx + qy*qy + qz*qz);
  qr *= qinv; qx *= qinv; qy *= qinv; qz *= qinv;

  float R[3][3];
  R[0][0] = 1.f - 2.f*(qy*qy + qz*qz); R[0][1] = 2.f*(qx*qy - qr*qz); R[0][2] = 2.f*(qx*qz + qr*qy);
  R[1][0] = 2.f*(qx*qy + qr*qz); R[1][1] = 1.f - 2.f*(qx*qx + qz*qz); R[1][2] = 2.f*(qy*qz - qr*qx);
  R[2][0] = 2.f*(qx*qz - qr*qy); R[2][1] = 2.f*(qy*qz + qr*qx); R[2][2] = 1.f - 2.f*(qx*qx + qy*qy);

  float s0 = scales[i*3+0], s1 = scales[i*3+1], s2 = scales[i*3+2];
  float RS[3][3];
  #pragma unroll
  for (int a = 0; a < 3; ++a) { RS[a][0] = R[a][0]*s0; RS[a][1] = R[a][1]*s1; RS[a][2] = R[a][2]*s2; }
  float C3[3][3];
  #pragma unroll
  for (int a = 0; a < 3; ++a)
    #pragma unroll
    for (int b = 0; b < 3; ++b)
      C3[a][b] = RS[a][0]*RS[b][0] + RS[a][1]*RS[b][1] + RS[a][2]*RS[b][2];

  float m0 = means[i*3+0], m1 = means[i*3+1], m2 = means[i*3+2];
  float V[16], P[16];
  #pragma unroll
  for (int k = 0; k < 16; ++k) { V[k] = vm[k]; P[k] = pm[k]; }

  // row-vector convention: pv = po @ V ; ph = pv @ P
  float po[4] = {m0, m1, m2, 1.0f};
  float pv[4], ph[4];
  #pragma unroll
  for (int l = 0; l < 4; ++l)
    pv[l] = po[0]*V[0*4+l] + po[1]*V[1*4+l] + po[2]*V[2*4+l] + po[3]*V[3*4+l];
  #pragma unroll
  for (int k = 0; k < 4; ++k)
    ph[k] = pv[0]*P[0*4+k] + pv[1]*P[1*4+k] + pv[2]*P[2*4+k] + pv[3]*P[3*4+k];
  float pw  = ph[3] + 1e-6f;
  float pp0 = ph[0] / pw;
  float pp1 = ph[1] / pw;
  depth[i] = pv[2];

  // t = means @ Rv^T + Tv  (Rv = V[:3,:3], Tv = V[:3,3])
  float t0 = m0*V[0] + m1*V[1] + m2*V[2]  + V[3];
  float t1 = m0*V[4] + m1*V[5] + m2*V[6]  + V[7];
  float t2 = m0*V[8] + m1*V[9] + m2*V[10] + V[11];
  float lim = 1.3f * TANF_C;
  float tx = clampf(t0 / t2, -lim, lim) * t2;
  float ty = clampf(t1 / t2, -lim, lim) * t2;
  float iz = 1.0f / (t2 + 1e-6f);

  float J00 = FOCAL_C * iz, J02 = -FOCAL_C * tx * iz * iz;
  float J11 = FOCAL_C * iz, J12 = -FOCAL_C * ty * iz * iz;

  // Wm = Rv^T ; cov_cam = Wm C3 Wm^T
  float Wm[3][3];
  #pragma unroll
  for (int a = 0; a < 3; ++a)
    #pragma unroll
    for (int b = 0; b < 3; ++b)
      Wm[a][b] = V[b*4 + a];
  float M1[3][3];
  #pragma unroll
  for (int a = 0; a < 3; ++a)
    #pragma unroll
    for (int k = 0; k < 3; ++k)
      M1[a][k] = Wm[a][0]*C3[0][k] + Wm[a][1]*C3[1][k] + Wm[a][2]*C3[2][k];
  float CC[3][3];
  #pragma unroll
  for (int a = 0; a < 3; ++a)
    #pragma unroll
    for (int l = 0; l < 3; ++l)
      CC[a][l] = M1[a][0]*Wm[l][0] + M1[a][1]*Wm[l][1] + M1[a][2]*Wm[l][2];

  // cov2d = J CC J^T + 0.3 I  (J row0 = (J00,0,J02), row1 = (0,J11,J12))
  float u0 = CC[0][0]*J00 + CC[0][2]*J02;
  float u2 = CC[2][0]*J00 + CC[2][2]*J02;
  float w0c = CC[0][1]*J11 + CC[0][2]*J12;
  float w1c = CC[1][1]*J11 + CC[1][2]*J12;
  float w2c = CC[2][1]*J11 + CC[2][2]*J12;
  float a2 = J00*u0  + J02*u2  + 0.3f;
  float b2 = J00*w0c + J02*w2c;
  float d2 = J11*w1c + J12*w2c + 0.3f;

  // radii from stop-gradient cov2d
  float det = a2*d2 - b2*b2;
  float mid = 0.5f * (a2 + d2);
  float sq  = sqrtf(fmaxf(mid*mid - det, 0.1f));
  float radii = 3.0f * ceilf(sqrtf(fmaxf(mid + sq, mid - sq)));
  radiiOut[i] = radii;

  float mx2 = ((pp0 + 1.0f) * (float)W_IMG - 1.0f) * 0.5f;
  float my2 = ((pp1 + 1.0f) * (float)H_IMG - 1.0f) * 0.5f;

  rminU[i*2+0] = clampf(mx2 - radii, 0.0f, (float)W_IMG - 1.0f);
  rminU[i*2+1] = clampf(my2 - radii, 0.0f, (float)H_IMG - 1.0f);
  rmaxU[i*2+0] = clampf(mx2 + radii, 0.0f, (float)W_IMG - 1.0f);
  rmaxU[i*2+1] = clampf(my2 + radii, 0.0f, (float)H_IMG - 1.0f);

  // Sinv of (cov2d + 1e-6 I)
  float ai = a2 + 1e-6f, di = d2 + 1e-6f;
  float idet = 1.0f / (ai*di - b2*b2);
  float S00 = di * idet, S01 = -b2 * idet, S11 = ai * idet;

  // Q(x,y) = f . c  with f = [x^2, y^2, xy, x, y, 1, 0, 0]
  coefU[i*8+0] = S00;
  coefU[i*8+1] = S11;
  coefU[i*8+2] = 2.0f * S01;
  coefU[i*8+3] = -2.0f * (S00*mx2 + S01*my2);
  coefU[i*8+4] = -2.0f * (S11*my2 + S01*mx2);
  coefU[i*8+5] = S00*mx2*mx2 + 2.0f*S01*mx2*my2 + S11*my2*my2;
  coefU[i*8+6] = 0.0f;
  coefU[i*8+7] = 0.0f;

  opU[i] = opac[i];
  colU[i*3+0] = fmaxf(cols[i*3+0] + 0.5f, 0.0f);
  colU[i*3+1] = fmaxf(cols[i*3+1] + 0.5f, 0.0f);
  colU[i*3+2] = fmaxf(cols[i*3+2] + 0.5f, 0.0f);
}

// ---------------------------------------------------------------------------
// Kernel 2: stable O(N^2) rank sort by depth, scatter params into sorted order.
// ---------------------------------------------------------------------------
__global__ void gs_sort_scatter(const float* __restrict__ depth,
                                const float* __restrict__ coefU,
                                const float* __restrict__ opU,
                                const float* __restrict__ colU,
                                const float* __restrict__ rminU,
                                const float* __restrict__ rmaxU,
                                float* __restrict__ coefS,
                                float* __restrict__ opS,
                                float* __restrict__ colS,
                                float* __restrict__ rminS,
                                float* __restrict__ rmaxS)
{
  int i = blockIdx.x * blockDim.x + threadIdx.x;
  if (i >= N_G) return;
  float d = depth[i];
  int rank = 0;
  for (int j = 0; j < N_G; ++j) {
    float dj = depth[j];
    rank += (dj < d) || (dj == d && j < i);
  }
  #pragma unroll
  for (int k = 0; k < 8; ++k) coefS[rank*8 + k] = coefU[i*8 + k];
  opS[rank] = opU[i];
  colS[rank*3+0] = colU[i*3+0];
  colS[rank*3+1] = colU[i*3+1];
  colS[rank*3+2] = colU[i*3+2];
  rminS[rank*2+0] = rminU[i*2+0];
  rminS[rank*2+1] = rminU[i*2+1];
  rmaxS[rank*2+0] = rmaxU[i*2+0];
  rmaxS[rank*2+1] = rmaxU[i*2+1];
}

// ---------------------------------------------------------------------------
// Kernel 3: per-tile render. 1 block per 25x25 tile, 640 threads (20 waves).
// Per 16-Gaussian chunk: V_WMMA_F32_16X16X4_F32 computes the 640x16 exponent
// block (pixel features x Gaussian coefficients), staged through LDS, then
// each thread blends its pixel sequentially in depth order. Chunk masks are
// packed into one bitmask word (readfirstlane -> uniform scalar branches, so
// WMMA is never executed under a partial EXEC mask) and empty chunks are
// skipped entirely.
// ---------------------------------------------------------------------------
__global__ __launch_bounds__(640, 1) void gs_render(const float* __restrict__ coefS,
                                                    const float* __restrict__ opS,
                                                    const float* __restrict__ colS,
                                                    const float* __restrict__ rminS,
                                                    const float* __restrict__ rmaxS,
                                                    float* __restrict__ outR,
                                                    float* __restrict__ outA)
{
  __shared__ float Qbuf[MPAD * 16];          // 40 KB exponent staging
  __shared__ float bcoef[16 * 8];            // chunk coefficients
  __shared__ float s_op[16], s_cr[16], s_cg[16], s_cb[16];
  __shared__ int   s_maskbits;               // 16 tile-level masks, packed
  __shared__ int   s_any;

  const int tid   = threadIdx.x;
  const int lane  = tid & 31;
  const int wave  = tid >> 5;
  const int ncol  = lane & 15;               // N index (Gaussian in chunk)
  const bool hi   = lane >= 16;
  const int khalf = hi ? 2 : 0;              // K pair held by this half-wave
  const int rhalf = hi ? 8 : 0;              // D-matrix row offset
  const float h0  = (float)((blockIdx.x >> 3) * TILE_SZ);
  const float w0  = (float)((blockIdx.x & 7) * TILE_SZ);

  // A-fragments: pixel features, fixed for the whole kernel. 2 M-tiles/wave.
  float afrag[2][4];
  #pragma unroll
  for (int mt = 0; mt < 2; ++mt) {
    int p = (wave * 2 + mt) * 16 + (lane & 15);   // M row = pixel-in-tile
    float x = 0.f, y = 0.f, one = 0.f;
    if (p < NPIX) {
      x = w0 + (float)(p % TILE_SZ);
      y = h0 + (float)(p / TILE_SZ);
      one = 1.0f;
    }
    // f = [x^2, y^2, xy, x, y, 1, 0, 0]; V0/V1 hold K=khalf / khalf+1
    afrag[mt][0] = hi ? x * y : x * x;   // kstep0 V0 : K2 | K0
    afrag[mt][1] = hi ? x     : y * y;   // kstep0 V1 : K3 | K1
    afrag[mt][2] = hi ? 0.0f  : y;       // kstep1 V0 : K6 | K4
    afrag[mt][3] = hi ? 0.0f  : one;     // kstep1 V1 : K7 | K5
  }

  float T = 1.0f, aR = 0.f, aG = 0.f, aB = 0.f, aA = 0.f;
  if (tid == 0) s_any = 0;
  __syncthreads();

  for (int ch = 0; ch < NCHUNK; ++ch) {
    // --- phase 1: stage chunk params into LDS, build packed mask word ---
    if (tid < 128) {
      bcoef[tid] = coefS[ch * 128 + tid];
    }
    if (tid >= 128 && tid < 160) {           // wave 4: masks via ballot
      int n = tid - 128;                     // 0..31 (only n<16 meaningful)
      int gm = ch * 16 + (n & 15);
      float tlx = fmaxf(rminS[gm*2+0], w0);
      float tly = fmaxf(rminS[gm*2+1], h0);
      float brx = fminf(rmaxS[gm*2+0], w0 + (float)TILE_SZ - 1.0f);
      float bry = fminf(rmaxS[gm*2+1], h0 + (float)TILE_SZ - 1.0f);
      bool m = (n < 16) && (brx > tlx) && (bry > tly);
      unsigned long long bal = __ballot(m);
      if (n == 0) {
        int mb = (int)(bal & 0xFFFFu);
        s_maskbits = mb;
        if (mb) s_any = 1;
      }
      if (n < 16) {
        int g = ch * 16 + n;
        s_op[n] = opS[g];
        s_cr[n] = colS[g*3+0];
        s_cg[n] = colS[g*3+1];
        s_cb[n] = colS[g*3+2];
      }
    }
    if (tid == 160 && ch + 1 < NCHUNK) {
      __builtin_prefetch(coefS + (ch + 1) * 128, 0, 1);  // global_prefetch_b8
    }
    __syncthreads();

    // uniform scalar chunk mask (forces s_cbranch, keeps EXEC all-1s for WMMA)
    const int mbits = __builtin_amdgcn_readfirstlane(s_maskbits);

    if (mbits != 0) {
      // --- phase 2: WMMA exponent GEMM into Qbuf ---
      v2f b0 = { bcoef[ncol*8 + khalf],     bcoef[ncol*8 + khalf + 1] };
      v2f b1 = { bcoef[ncol*8 + 4 + khalf], bcoef[ncol*8 + 5 + khalf] };
      #pragma unroll
      for (int mt = 0; mt < 2; ++mt) {
        v2f a0 = { afrag[mt][0], afrag[mt][1] };
        v2f a1 = { afrag[mt][2], afrag[mt][3] };
        v8f acc = {0.f, 0.f, 0.f, 0.f, 0.f, 0.f, 0.f, 0.f};
        acc = __builtin_amdgcn_wmma_f32_16x16x4_f32(false, a0, false, b0,
                                                    (short)0, acc, false, false);
        acc = __builtin_amdgcn_wmma_f32_16x16x4_f32(false, a1, false, b1,
                                                    (short)0, acc, false, false);
        int pbase = (wave * 2 + mt) * 16 + rhalf;
        #pragma unroll
        for (int r = 0; r < 8; ++r)
          Qbuf[(pbase + r) * 16 + ncol] = acc[r];
      }
    }
    __syncthreads();

    if (mbits != 0 && tid < NPIX) {
      // --- phase 3a: 16 independent alphas (exp's pipeline on trans unit) ---
      float alpha[16];
      #pragma unroll
      for (int n = 0; n < 16; ++n) {
        float q = Qbuf[tid * 16 + n];
        float a = s_op[n] * __expf(-0.5f * q);
        alpha[n] = ((mbits >> n) & 1) ? a : 0.0f;
      }
      // --- phase 3b: short serial transmittance chain, branchless ---
      #pragma unroll
      for (int n = 0; n < 16; ++n) {
        float wgt = T * alpha[n];            // 0 when masked off
        aR += wgt * s_cr[n];
        aG += wgt * s_cg[n];
        aB += wgt * s_cb[n];
        aA += wgt;
        float f = ((mbits >> n) & 1) ? (1.0f - alpha[n] + 1e-10f) : 1.0f;
        T *= f;
      }
    }
    __syncthreads();
  }

  if (tid < NPIX) {
    int X = (int)w0 + tid % TILE_SZ;
    int Y = (int)h0 + tid / TILE_SZ;
    bool has = (s_any != 0);
    int pix = Y * W_IMG + X;
    outR[pix*3+0] = has ? aR : 1.0f;
    outR[pix*3+1] = has ? aG : 1.0f;
    outR[pix*3+2] = has ? aB : 1.0f;
    outA[pix]     = has ? aA : 0.0f;
  }
}

// ---------------------------------------------------------------------------
extern "C" void kernel_launch(void* const* d_in, const int* in_sizes, int n_in,
                              void* d_out, int out_size, void* d_ws, size_t ws_size,
                              hipStream_t stream)
{
  const float* means  = (const float*)d_in[0];
  const float* scales = (const float*)d_in[1];
  const float* rots   = (const float*)d_in[2];
  const float* opac   = (const float*)d_in[3];
  const float* cols   = (const float*)d_in[4];
  const float* vm     = (const float*)d_in[5];
  const float* pm     = (const float*)d_in[6];

  float* out      = (float*)d_out;
  float* outR     = out;                         // (H,W,3)
  float* outA     = out + H_IMG * W_IMG * 3;     // (H,W,1)
  float* radiiOut = out + H_IMG * W_IMG * 4;     // (N)

  float* ws     = (float*)d_ws;
  float* depth  = ws;
  float* coefU  = depth  + N_G;
  float* opU    = coefU  + N_G * 8;
  float* colU   = opU    + N_G;
  float* rminU  = colU   + N_G * 3;
  float* rmaxU  = rminU  + N_G * 2;
  float* coefSrt = rmaxU  + N_G * 2;
  float* opSrt   = coefSrt + N_G * 8;
  float* colSrt  = opSrt   + N_G;
  float* rminSrt = colSrt  + N_G * 3;
  float* rmaxSrt = rminSrt + N_G * 2;

  gs_preprocess<<<4, 256, 0, stream>>>(means, scales, rots, opac, cols, vm, pm,
                                       depth, coefU, opU, colU, rminU, rmaxU,
                                       radiiOut);
  gs_sort_scatter<<<4, 256, 0, stream>>>(depth, coefU, opU, colU, rminU, rmaxU,
                                         coefSrt, opSrt, colSrt, rminSrt, rmaxSrt);
  gs_render<<<64, 640, 0, stream>>>(coefSrt, opSrt, colSrt, rminSrt, rmaxSrt,
                                    outR, outA);
}